// HPR_WN_72353019068521
// MI455X (gfx1250) — compile-verified
//
#include <hip/hip_runtime.h>
#include <hip/hip_bf16.h>
#include <math.h>

#define CC   0.01f
#define SCC  0.1f       /* sqrt(C) */
#define MINN 1e-5f
#define MAXN 9.99f      /* (1-0.001)/sqrt(C) */
#define DD   1024
#define KC   5          /* classes */
#define QQ   75         /* queries */
#define NAUG 40

typedef __attribute__((ext_vector_type(2))) float v2f;
typedef __attribute__((ext_vector_type(8))) float v8f;

__device__ __forceinline__ float bred(float v, float* sA, int t) {
    sA[t] = v; __syncthreads();
    for (int s = 128; s > 0; s >>= 1) { if (t < s) sA[t] += sA[t + s]; __syncthreads(); }
    float r = sA[0]; __syncthreads(); return r;
}

__device__ __forceinline__ void bred2(float va, float vb, float* sA, float* sB,
                                      int t, float* ra, float* rb) {
    sA[t] = va; sB[t] = vb; __syncthreads();
    for (int s = 128; s > 0; s >>= 1) {
        if (t < s) { sA[t] += sA[t + s]; sB[t] += sB[t + s]; }
        __syncthreads();
    }
    *ra = sA[0]; *rb = sB[0]; __syncthreads();
}

/* ---- zero-fill (padding rows, run every call) ---- */
__global__ void k_zero(float* __restrict__ p, int n) {
    int i = blockIdx.x * blockDim.x + threadIdx.x;
    if (i < n) p[i] = 0.f;
}

/* ---- expmap0 + project for all 100 feat rows ---- */
__global__ void k_map(const float* __restrict__ feat,
                      float* __restrict__ sp, float* __restrict__ qp,
                      float* __restrict__ spn, float* __restrict__ qpn) {
    __shared__ float sA[256];
    int r = blockIdx.x, t = threadIdx.x;
    const float* u = feat + (size_t)r * DD + t * 4;
    float x0 = u[0], x1 = u[1], x2 = u[2], x3 = u[3];
    float u2 = bred(x0*x0 + x1*x1 + x2*x2 + x3*x3, sA, t);
    float nu = sqrtf(u2);
    float un = fmaxf(nu, MINN);
    float tt = tanhf(SCC * un) / (SCC * un);
    float pn = tt * nu;                 /* ||expmap0(u)|| */
    float f  = tt;
    if (pn > MAXN) { f *= MAXN / pn; pn = MAXN; }
    float p0 = x0*f, p1 = x1*f, p2 = x2*f, p3 = x3*f;
    if (r < 25) {
        float* d = sp + (size_t)r * DD + t * 4;
        d[0]=p0; d[1]=p1; d[2]=p2; d[3]=p3;
        if (t == 0) spn[r] = pn * pn;
    } else {
        float* d = qp + (size_t)(r - 25) * DD + t * 4;
        d[0]=p0; d[1]=p1; d[2]=p2; d[3]=p3;
        if (t == 0) qpn[r - 25] = pn * pn;
    }
}

/* ---- per-class Poincare mean of 5 supports -> proto_p ---- */
__global__ void k_proto(const float* __restrict__ sp, const float* __restrict__ spn,
                        float* __restrict__ proto, float* __restrict__ pnrm) {
    __shared__ float sA[256];
    int c = blockIdx.x, t = threadIdx.x, d = t * 4;
    float a0=0,a1=0,a2=0,a3=0, wsum=0;
    for (int i = 0; i < 5; i++) {
        float n2 = spn[c*5 + i];
        float a  = 2.f / (1.f + CC * n2);
        float k2 = a * a * n2;
        float lam = 1.f / sqrtf(fmaxf(1.f - CC * k2, MINN));
        wsum += lam;
        float cf = lam * a;
        const float* p = sp + (size_t)(c*5 + i) * DD + d;
        a0 += cf*p[0]; a1 += cf*p[1]; a2 += cf*p[2]; a3 += cf*p[3];
    }
    a0/=wsum; a1/=wsum; a2/=wsum; a3/=wsum;
    float mk2 = bred(a0*a0 + a1*a1 + a2*a2 + a3*a3, sA, t);
    float den = 1.f + sqrtf(fmaxf(1.f - CC * mk2, 0.f));
    float* o = proto + (size_t)c * DD + d;
    o[0]=a0/den; o[1]=a1/den; o[2]=a2/den; o[3]=a3/den;
    if (t == 0) pnrm[c] = mk2 / (den * den);
}

/* ---- WMMA GEMM: O[80x16] = Q(80x1024, rows>=75 zero) * P(16x1024, rows>=5 zero)^T ----
 * One block, 5 wave32s; wave w owns output row-tile [16w, 16w+16).
 * f32 16x16x4 WMMA: A frag = {K+0,K+1} lanes 0-15 / {K+2,K+3} lanes 16-31; B mirrors. */
__global__ void k_gemm(const float* __restrict__ Q, const float* __restrict__ P,
                       float* __restrict__ O) {
    int lane = threadIdx.x & 31;
    int wave = threadIdx.x >> 5;
    int m0   = wave * 16;
    int row  = m0 + (lane & 15);
    int col  = lane & 15;
    int koff = (lane >> 4) * 2;          /* lanes 16-31 hold K+2,K+3 */
    const float* qrow = Q + (size_t)row * DD + koff;
    const float* prow = P + (size_t)col * DD + koff;
    v8f acc = {};
    for (int k = 0; k < DD; k += 32) {
        __builtin_prefetch(qrow + k + 32, 0, 3);
        __builtin_prefetch(prow + k + 32, 0, 3);
#pragma unroll
        for (int u = 0; u < 8; u++) {
            v2f a = *(const v2f*)(qrow + k + u * 4);
            v2f b = *(const v2f*)(prow + k + u * 4);
            acc = __builtin_amdgcn_wmma_f32_16x16x4_f32(
                false, a, false, b, (short)0, acc, false, false);
        }
    }
#pragma unroll
    for (int r = 0; r < 8; r++) {
        int mr = m0 + r + ((lane >> 4) << 3);
        O[mr * 16 + col] = acc[r];
    }
}

__device__ __forceinline__ float hyp_dist(float x2, float y2, float xy) {
    float a = 1.f - 2.f*CC*xy + CC*y2;
    float b = 1.f - CC*x2;
    float num2 = a*a*x2 + b*b*y2 - 2.f*a*b*xy;
    float den  = fmaxf(1.f - 2.f*CC*xy + CC*CC*x2*y2, MINN);
    float nr   = sqrtf(fmaxf(num2, 0.f)) / den;
    float z    = SCC * nr;
    z = fminf(fmaxf(z, -1.f + 1e-5f), 1.f - 1e-5f);
    return (1.f / SCC) * logf((1.f + z) / (1.f - z));   /* 2/sc * artanh */
}

/* ---- dist matrix + argmax(-dist) ---- */
__global__ void k_dist(const float* __restrict__ xy, const float* __restrict__ qpn,
                       const float* __restrict__ pn, float* __restrict__ dist,
                       int* __restrict__ pred) {
    int i = threadIdx.x;
    if (i >= QQ) return;
    float x2 = qpn[i];
    float best = 1e30f; int bi = 0;
    for (int c = 0; c < KC; c++) {
        float dv = hyp_dist(x2, pn[c], xy[i * 16 + c]);
        dist[i * 5 + c] = dv;
        if (dv < best) { best = dv; bi = c; }
    }
    pred[i] = bi;
}

/* ---- per-class: masked means, std, 40 augmented expmaps, new prototype ---- */
__global__ void k_aug(const float* __restrict__ sp, const float* __restrict__ spn,
                      const float* __restrict__ qp, const float* __restrict__ qpn,
                      const float* __restrict__ feat, const int* __restrict__ pred,
                      const float* __restrict__ noise,
                      float* __restrict__ pnew, float* __restrict__ pnn) {
    __shared__ float sA[256];
    __shared__ float sB[256];
    __shared__ float shW[80];
    __shared__ float shK[80];
    int c = blockIdx.x, t = threadIdx.x, d = t * 4;

    if (t < 80) {
        float m, n2;
        if (t < 5) { m = 1.f; n2 = spn[c*5 + t]; }
        else { int j = t - 5; m = (pred[j] == c) ? 1.f : 0.f; n2 = qpn[j]; }
        float a  = 2.f / (1.f + CC * n2);
        float k2 = a * a * n2;
        float lam = 1.f / sqrtf(fmaxf(1.f - CC * k2, MINN));
        shW[t] = lam * m;
        shK[t] = lam * m * a;
    }
    __syncthreads();

    float wsum = 0.f, cnt = 0.f;
    for (int i = 0; i < 80; i++) { wsum += shW[i]; if (shW[i] > 0.f) cnt += 1.f; }

    /* masked Poincare mean -> s_p */
    float a0=0,a1=0,a2=0,a3=0;
    for (int i = 0; i < 80; i++) {
        float cf = shK[i];
        if (cf != 0.f) {
            const float* p = (i < 5) ? sp + (size_t)(c*5 + i) * DD + d
                                     : qp + (size_t)(i - 5) * DD + d;
            a0 += cf*p[0]; a1 += cf*p[1]; a2 += cf*p[2]; a3 += cf*p[3];
        }
    }
    a0/=wsum; a1/=wsum; a2/=wsum; a3/=wsum;
    float mk2 = bred(a0*a0 + a1*a1 + a2*a2 + a3*a3, sA, t);
    float denp = 1.f + sqrtf(fmaxf(1.f - CC * mk2, 0.f));
    float s0=a0/denp, s1=a1/denp, s2=a2/denp, s3=a3/denp;
    float sp2 = mk2 / (denp * denp);

    /* masked Euclidean mean/std over feat */
    float sm0=0,sm1=0,sm2=0,sm3=0, sq0=0,sq1=0,sq2=0,sq3=0;
    for (int i = 0; i < 80; i++) {
        if (shW[i] > 0.f) {
            const float* e = (i < 5) ? feat + (size_t)(c*5 + i) * DD + d
                                     : feat + (size_t)(25 + i - 5) * DD + d;
            float e0=e[0],e1=e[1],e2=e[2],e3=e[3];
            sm0+=e0; sm1+=e1; sm2+=e2; sm3+=e3;
            sq0+=e0*e0; sq1+=e1*e1; sq2+=e2*e2; sq3+=e3*e3;
        }
    }
    float mu0=sm0/cnt, mu1=sm1/cnt, mu2=sm2/cnt, mu3=sm3/cnt;
    float inv = 1.f / (cnt - 1.f);
    float st0 = sqrtf(fmaxf((sq0 - cnt*mu0*mu0) * inv, 0.f));
    float st1 = sqrtf(fmaxf((sq1 - cnt*mu1*mu1) * inv, 0.f));
    float st2 = sqrtf(fmaxf((sq2 - cnt*mu2*mu2) * inv, 0.f));
    float st3 = sqrtf(fmaxf((sq3 - cnt*mu3*mu3) * inv, 0.f));

    float lam_sp = 2.f / fmaxf(1.f - CC * sp2, MINN);
    float fpt    = 1.f - CC * sp2;       /* ptransp0 factor */

    /* final mean accumulation: supports first (mask 1 == shW/shK for i<5) */
    float b0=0,b1=0,b2=0,b3=0, wsum2=0;
    for (int i = 0; i < 5; i++) {
        wsum2 += shW[i];
        float cf = shK[i];
        const float* p = sp + (size_t)(c*5 + i) * DD + d;
        b0 += cf*p[0]; b1 += cf*p[1]; b2 += cf*p[2]; b3 += cf*p[3];
    }

    const float* nz = noise + (size_t)c * NAUG * DD;
    for (int j = 0; j < NAUG; j++) {
        const float* n = nz + (size_t)j * DD + d;
        float p0 = n[0]*st0*fpt, p1 = n[1]*st1*fpt, p2 = n[2]*st2*fpt, p3 = n[3]*st3*fpt;
        float un2, dxy;
        bred2(p0*p0 + p1*p1 + p2*p2 + p3*p3,
              s0*p0 + s1*p1 + s2*p2 + s3*p3, sA, sB, t, &un2, &dxy);
        float un = fmaxf(sqrtf(un2), MINN);
        float sc2 = tanhf(SCC * lam_sp * un * 0.5f) / (SCC * un);
        float y2  = sc2 * sc2 * un2;
        float xyv = sc2 * dxy;
        float A = 1.f + 2.f*CC*xyv + CC*y2;
        float B = (1.f - CC*sp2) * sc2;
        float den = fmaxf(1.f + 2.f*CC*xyv + CC*CC*sp2*y2, MINN);
        float e0 = (A*s0 + B*p0)/den, e1 = (A*s1 + B*p1)/den;
        float e2 = (A*s2 + B*p2)/den, e3 = (A*s3 + B*p3)/den;
        float ex2 = bred(e0*e0 + e1*e1 + e2*e2 + e3*e3, sA, t);
        float aa = 2.f / (1.f + CC * ex2);
        float k2 = aa * aa * ex2;
        float lam = 1.f / sqrtf(fmaxf(1.f - CC * k2, MINN));
        wsum2 += lam;
        float cf = lam * aa;
        b0 += cf*e0; b1 += cf*e1; b2 += cf*e2; b3 += cf*e3;
    }
    b0/=wsum2; b1/=wsum2; b2/=wsum2; b3/=wsum2;
    float mk2b = bred(b0*b0 + b1*b1 + b2*b2 + b3*b3, sA, t);
    float den2 = 1.f + sqrtf(fmaxf(1.f - CC * mk2b, 0.f));
    float* o = pnew + (size_t)c * DD + d;
    o[0]=b0/den2; o[1]=b1/den2; o[2]=b2/den2; o[3]=b3/den2;
    if (t == 0) pnn[c] = mk2b / (den2 * den2);
}

/* ---- softmax(-dist_new) + double log-softmax loss ---- */
__global__ void k_final(const float* __restrict__ xyn, const float* __restrict__ qpn,
                        const float* __restrict__ pnn, const float* __restrict__ dist,
                        const int* __restrict__ label, float* __restrict__ out) {
    __shared__ float sA[128];
    int i = threadIdx.x;
    float contrib = 0.f;
    if (i < QQ) {
        float x2 = qpn[i];
        float z[KC], m = -1e30f;
        for (int c = 0; c < KC; c++) {
            z[c] = -hyp_dist(x2, pnn[c], xyn[i * 16 + c]);
            m = fmaxf(m, z[c]);
        }
        float s = 0.f;
        for (int c = 0; c < KC; c++) s += expf(z[c] - m);
        for (int c = 0; c < KC; c++) out[i * KC + c] = expf(z[c] - m) / s;

        float w[KC], m1 = -1e30f;
        for (int c = 0; c < KC; c++) { w[c] = -dist[i * KC + c]; m1 = fmaxf(m1, w[c]); }
        float s1 = 0.f;
        for (int c = 0; c < KC; c++) s1 += expf(w[c] - m1);
        float l1 = m1 + logf(s1);
        float m2 = -1e30f;
        for (int c = 0; c < KC; c++) { w[c] -= l1; m2 = fmaxf(m2, w[c]); }
        float s2 = 0.f;
        for (int c = 0; c < KC; c++) s2 += expf(w[c] - m2);
        float l2 = m2 + logf(s2);
        int lb = label[QQ + i];          /* label[1][i] */
        contrib = -(w[lb] - l2) / (float)QQ;
    }
    sA[i] = contrib; __syncthreads();
    for (int s = 64; s > 0; s >>= 1) { if (i < s) sA[i] += sA[i + s]; __syncthreads(); }
    if (i == 0) out[QQ * KC] = sA[0];
}

extern "C" void kernel_launch(void* const* d_in, const int* in_sizes, int n_in,
                              void* d_out, int out_size, void* d_ws, size_t ws_size,
                              hipStream_t stream) {
    const float* feat  = (const float*)d_in[0];
    const int*   label = (const int*)d_in[1];
    const float* noise = (const float*)d_in[2];

    float* ws    = (float*)d_ws;
    float* qp    = ws;                  /* 80*1024 = 81920 (rows 75-79 zero) */
    float* sp    = ws + 81920;          /* 25*1024 = 25600 */
    float* qpn   = ws + 107520;         /* 80  */
    float* spn   = ws + 107600;         /* 32  */
    float* proto = ws + 107632;         /* 16*1024 = 16384 (rows 5-15 zero) */
    float* pnrm  = ws + 124016;         /* 16  */
    float* xyb   = ws + 124032;         /* 80*16 = 1280 */
    float* distb = ws + 125312;         /* 384 */
    int*   pred  = (int*)(ws + 125696); /* 80  */
    float* pnew  = ws + 125776;         /* 16*1024 = 16384 (rows 5-15 zero) */
    float* pnn   = ws + 142160;         /* 16  */
    float* xyn   = ws + 142176;         /* 1280 */

    /* zero the padding rows every call (graph-safe, deterministic) */
    k_zero<<<(5 * DD + 255) / 256, 256, 0, stream>>>(qp + 75 * DD, 5 * DD);
    k_zero<<<(11 * DD + 255) / 256, 256, 0, stream>>>(proto + 5 * DD, 11 * DD);
    k_zero<<<(11 * DD + 255) / 256, 256, 0, stream>>>(pnew + 5 * DD, 11 * DD);

    k_map  <<<100, 256, 0, stream>>>(feat, sp, qp, spn, qpn);
    k_proto<<<KC,  256, 0, stream>>>(sp, spn, proto, pnrm);
    k_gemm <<<1,   160, 0, stream>>>(qp, proto, xyb);
    k_dist <<<1,   128, 0, stream>>>(xyb, qpn, pnrm, distb, pred);
    k_aug  <<<KC,  256, 0, stream>>>(sp, spn, qp, qpn, feat, pred, noise, pnew, pnn);
    k_gemm <<<1,   160, 0, stream>>>(qp, pnew, xyn);
    k_final<<<1,   128, 0, stream>>>(xyn, qpn, pnn, distb, label, (float*)d_out);
}